// MyLayer_25915832664834
// MI455X (gfx1250) — compile-verified
//
#include <hip/hip_runtime.h>
#include <hip/hip_bf16.h>

// MI455X / gfx1250, wave32.
// Problem: BATCH=8192, D=512, F=64, K=64.
//   W[i,j] = dot_k( kernel[i, fd[j], :], kernel[j, fd[i], :] ), strictly-upper masked.
//   out[b] = x_b^T Wm x_b  ==  rowdot( X, X @ Wm^T )   (GEMM M=8192,N=512,K=512)
// All data L2-resident; compute-bound on the GEMM -> fp32 WMMA 16x16x4.

typedef __attribute__((ext_vector_type(2))) float v2f;
typedef __attribute__((ext_vector_type(8))) float v8f;

#define BATCH 8192
#define DDIM  512
#define FDIM  64
#define KDIM  64

// ---------------------------------------------------------------------------
// Kernel 1: build masked W (512x512 f32) into workspace.
// One thread per (i,j). K=64 dot with float4 loads (row-contiguous).
// ---------------------------------------------------------------------------
__global__ __launch_bounds__(256) void ffm_build_w(
    const float* __restrict__ kern,     // [D, F, K]
    const int*   __restrict__ fd,       // [D]
    float*       __restrict__ Wm)       // [D, D]
{
    int tid = blockIdx.x * 256 + threadIdx.x;   // 0 .. 512*512-1
    int i = tid >> 9;
    int j = tid & (DDIM - 1);
    float w = 0.0f;
    if (j > i) {
        // Whole wave shares i -> fd[i] is a scalar load; fd[j] coalesced.
        const float4* a = (const float4*)(kern + (size_t)i * (FDIM * KDIM) + (size_t)fd[j] * KDIM);
        const float4* b = (const float4*)(kern + (size_t)j * (FDIM * KDIM) + (size_t)fd[i] * KDIM);
#pragma unroll
        for (int k = 0; k < KDIM / 4; ++k) {
            float4 av = a[k], bv = b[k];
            w += av.x * bv.x + av.y * bv.y + av.z * bv.z + av.w * bv.w;
        }
    }
    Wm[tid] = w;  // zeros on/below the diagonal implement the triu(k=1) mask
}

// ---------------------------------------------------------------------------
// Kernel 2: out[b] = sum_i x[b,i] * (sum_j Wm[i,j] x[b,j])  via fp32 WMMA.
//
// Block = 16 batch rows (b0), 8 waves. Wave w owns column tiles
// {w, 15-w, 16+w, 31-w}; tile ti only needs j >= 16*ti (strict upper mask),
// so every wave executes exactly 264 WMMAs -> perfect balance.
//
// i0 and the trip count are wave-uniform SGPRs (readfirstlane) -> scalar
// branch loop. The j-loop iterates in j-steps of 16 with a fully unrolled
// 4x(2 loads + WMMA) body, so the trip count (32 - ti) needs no remainder
// loop; the compiler clauses the 8 b64 loads and staggers s_wait_loadcnt
// across the 4 WMMAs.
//
// fp32 WMMA 16x16x4 VGPR layout (ISA 7.12.2):
//   A (16x4, M=batch, K=j): lane l<16 -> M=l, {v0,v1}={K0,K1};
//                           lane l>=16 -> M=l-16, {v0,v1}={K2,K3}
//   B (4x16, K=j, N=i)    : lane n<16 -> N=n, {v0,v1}={K0,K1}; halves K2,K3
//   C/D (16x16)           : vgpr r, lanes 0-15 -> M=r, N=lane;
//                           lanes 16-31 -> M=r+8, N=lane-16
// ---------------------------------------------------------------------------
__global__ __launch_bounds__(256) void ffm_quadform(
    const float* __restrict__ x,        // [BATCH, D]
    const float* __restrict__ Wm,       // [D, D]
    float*       __restrict__ out)      // [BATCH]
{
    const int b0   = blockIdx.x * 16;
    const int wave = __builtin_amdgcn_readfirstlane(threadIdx.x >> 5); // SGPR, 0..7
    const int lane = threadIdx.x & 31;
    const int half = lane >> 4;             // 0 or 1 (selects K pair / M+8)
    const int l    = lane & 15;             // M for A-loads, N for B-loads

    __shared__ float partial[8][16];

    float contrib[8];
#pragma unroll
    for (int r = 0; r < 8; ++r) contrib[r] = 0.0f;

    const int tiles[4] = { wave, 15 - wave, 16 + wave, 31 - wave };

#pragma unroll
    for (int t = 0; t < 4; ++t) {
        // Wave-uniform tile base / trip count -> scalar loop control.
        const int i0  = __builtin_amdgcn_readfirstlane(tiles[t] << 4);
        const int n16 = __builtin_amdgcn_readfirstlane((DDIM - i0) >> 4); // 32 - ti >= 1

        v8f acc = {0.f, 0.f, 0.f, 0.f, 0.f, 0.f, 0.f, 0.f};

        // Strictly-upper mask: start at j0 = i0 (diagonal tile holds zeros).
        const v2f* __restrict__ xp =
            (const v2f*)(x  + (size_t)(b0 + l) * DDIM + i0 + 2 * half);
        const v2f* __restrict__ wp =
            (const v2f*)(Wm + (size_t)(i0 + l) * DDIM + i0 + 2 * half);

        for (int it = 0; it < n16; ++it) {      // 16 j-values per iteration
#pragma unroll
            for (int u = 0; u < 4; ++u) {
                v2f a = xp[2 * u];              // 8B-aligned -> global_load_b64
                v2f b = wp[2 * u];
                acc = __builtin_amdgcn_wmma_f32_16x16x4_f32(
                    /*neg_a=*/false, a, /*neg_b=*/false, b,
                    /*c_mod=*/(short)0, acc, /*reuse_a=*/false, /*reuse_b=*/false);
            }
            xp += 8;                            // advance 16 floats
            wp += 8;
        }

        // acc[r] = Y[b0 + r + 8*half, i0 + l]; fold in x[b, i0+l] now (N-dep).
#pragma unroll
        for (int r = 0; r < 8; ++r) {
            float xv = x[(size_t)(b0 + r + 8 * half) * DDIM + i0 + l];
            contrib[r] += acc[r] * xv;
        }
    }

    // Reduce over the 16 N-lanes inside each half (xor masks 1,2,4,8 never
    // cross bit 4, so the two halves reduce independently).
#pragma unroll
    for (int r = 0; r < 8; ++r) {
        float v = contrib[r];
        v += __shfl_xor(v, 1, 32);
        v += __shfl_xor(v, 2, 32);
        v += __shfl_xor(v, 4, 32);
        v += __shfl_xor(v, 8, 32);
        if (l == 0) partial[wave][r + 8 * half] = v;   // M = r + 8*half
    }
    __syncthreads();

    // Cross-wave reduce; each block exclusively owns its 16 outputs.
    if (threadIdx.x < 16) {
        float s = 0.0f;
#pragma unroll
        for (int w = 0; w < 8; ++w) s += partial[w][threadIdx.x];
        out[b0 + threadIdx.x] = s;
    }
}

// ---------------------------------------------------------------------------
extern "C" void kernel_launch(void* const* d_in, const int* in_sizes, int n_in,
                              void* d_out, int out_size, void* d_ws, size_t ws_size,
                              hipStream_t stream) {
    const float* x    = (const float*)d_in[0];   // [8192, 512] f32
    const float* kern = (const float*)d_in[1];   // [512, 64, 64] f32
    const int*   fd   = (const int*)d_in[2];     // [512] i32
    float* out = (float*)d_out;                  // [8192] f32
    float* Wm  = (float*)d_ws;                   // 512*512*4 = 1 MB scratch

    ffm_build_w<<<(DDIM * DDIM) / 256, 256, 0, stream>>>(kern, fd, Wm);
    ffm_quadform<<<BATCH / 16, 256, 0, stream>>>(x, Wm, out);
}